// GCN_2044404433338
// MI455X (gfx1250) — compile-verified
//
#include <hip/hip_runtime.h>
#include <math.h>

typedef __attribute__((ext_vector_type(2))) float v2f;
typedef __attribute__((ext_vector_type(8))) float v8f;

#define F 128  // F_IN == HID == 128

static __device__ __forceinline__ void atomic_fadd(float* p, float v) {
    // relaxed, device scope -> hardware global_atomic_add_f32 (no CAS loop)
    __hip_atomic_fetch_add(p, v, __ATOMIC_RELAXED, __HIP_MEMORY_SCOPE_AGENT);
}

// ---------------- degree / normalization ----------------
__global__ void k_deg_init(unsigned* __restrict__ deg, int n) {
    int i = blockIdx.x * blockDim.x + threadIdx.x;
    if (i < n) deg[i] = 1u;  // self loop
}

__global__ void k_deg_count(const int* __restrict__ dst, unsigned* __restrict__ deg, int e) {
    int i = blockIdx.x * blockDim.x + threadIdx.x;
    if (i < e) atomicAdd(&deg[dst[i]], 1u);
}

__global__ void k_dinv(const unsigned* __restrict__ deg, float* __restrict__ dinv, int n) {
    int i = blockIdx.x * blockDim.x + threadIdx.x;
    if (i < n) dinv[i] = rsqrtf((float)deg[i]);  // deg >= 1 always
}

// ---------------- layer-1 GEMM: HW1 = X @ W1 via V_WMMA_F32_16X16X4_F32 ----------------
// Block = 256 threads = 8 waves. Block computes rows [16*bx, 16*bx+16) x all 128 cols.
// Wave w computes the 16-column slab [16w, 16w+16).  N is a multiple of 16 (100000 = 6250*16).
__global__ void __launch_bounds__(256) k_gemm1(const float* __restrict__ X,
                                               const float* __restrict__ W1,
                                               float* __restrict__ hw1) {
    const int wave = threadIdx.x >> 5;
    const int lane = threadIdx.x & 31;
    const int m    = lane & 15;
    const int koff = (lane >> 4) * 2;        // lanes 16-31 hold K+2 / K+3
    const int col  = wave * 16 + m;
    const int rowbase = blockIdx.x * 16;
    const float* __restrict__ xrow = X + (size_t)(rowbase + m) * F;

    v8f acc = {};
#pragma unroll
    for (int k0 = 0; k0 < F; k0 += 4) {
        v2f a;  // A 16x4: contiguous pair from this row
        a.x = xrow[k0 + koff];
        a.y = xrow[k0 + koff + 1];
        v2f b;  // B 4x16: two W1 rows, same column
        b.x = W1[(size_t)(k0 + koff)     * F + col];
        b.y = W1[(size_t)(k0 + koff + 1) * F + col];
        acc = __builtin_amdgcn_wmma_f32_16x16x4_f32(
            /*neg_a=*/false, a, /*neg_b=*/false, b,
            /*c_mod=*/(short)0, acc, /*reuse_a=*/false, /*reuse_b=*/false);
    }

    const int rowadd = (lane >> 4) * 8;      // C/D: vgpr j -> M=j (lanes 0-15), M=j+8 (16-31)
#pragma unroll
    for (int j = 0; j < 8; ++j)
        hw1[(size_t)(rowbase + j + rowadd) * F + col] = acc[j];
}

// ---------------- generic zero ----------------
__global__ void k_zero(float* __restrict__ p, long long n) {
    long long i = (long long)blockIdx.x * blockDim.x + threadIdx.x;
    if (i < n) p[i] = 0.0f;
}

// ---------------- layer-1 edge scatter: agg[dst] += norm * hw1[src] ----------------
// One wave per edge: 32 lanes each handle a float4 chunk (512B contiguous per row).
__global__ void k_scatter1(const float* __restrict__ hw1, const float* __restrict__ dinv,
                           const int* __restrict__ src, const int* __restrict__ dst,
                           float* __restrict__ agg, int e) {
    long long idx = (long long)blockIdx.x * blockDim.x + threadIdx.x;
    int ed = (int)(idx >> 5);
    if (ed >= e) return;
    int c = ((int)idx & 31) << 2;            // feature offset: 0,4,...,124
    int s = src[ed], d = dst[ed];
    float nrm = dinv[s] * dinv[d];
    const float4 v = *(const float4*)(hw1 + (size_t)s * F + c);
    float* o = agg + (size_t)d * F + c;
    atomic_fadd(o + 0, nrm * v.x);
    atomic_fadd(o + 1, nrm * v.y);
    atomic_fadd(o + 2, nrm * v.z);
    atomic_fadd(o + 3, nrm * v.w);
}

// ---------------- layer-1 epilogue: h = relu(agg + dinv^2*hw1 + b1), in place ----------------
__global__ void k_epi1(const float* __restrict__ hw1, const float* __restrict__ dinv,
                       const float* __restrict__ b1, float* __restrict__ h, long long total) {
    long long i = (long long)blockIdx.x * blockDim.x + threadIdx.x;
    if (i >= total) return;
    int row = (int)(i >> 7);
    int c   = (int)(i & (F - 1));
    float di = dinv[row];
    float v = h[i] + di * di * hw1[i] + b1[c];
    h[i] = fmaxf(v, 0.0f);
}

// ---------------- layer-2 mat-vec: hw2 = h @ W2  (128 -> 1) ----------------
__global__ void __launch_bounds__(256) k_matvec2(const float* __restrict__ h,
                                                 const float* __restrict__ W2,
                                                 float* __restrict__ hw2, int n) {
    int node = blockIdx.x * 8 + (threadIdx.x >> 5);
    int lane = threadIdx.x & 31;
    if (node >= n) return;
    const float4 hv = *(const float4*)(h + (size_t)node * F + lane * 4);
    const float4 wv = *(const float4*)(W2 + lane * 4);
    float p = hv.x * wv.x + hv.y * wv.y + hv.z * wv.z + hv.w * wv.w;
#pragma unroll
    for (int off = 16; off > 0; off >>= 1) p += __shfl_xor(p, off);
    if (lane == 0) hw2[node] = p;
}

// ---------------- layer-2 edge scatter (scalar) ----------------
__global__ void k_scatter2(const float* __restrict__ hw2, const float* __restrict__ dinv,
                           const int* __restrict__ src, const int* __restrict__ dst,
                           float* __restrict__ out, int e) {
    int i = blockIdx.x * blockDim.x + threadIdx.x;
    if (i >= e) return;
    int s = src[i], d = dst[i];
    atomic_fadd(&out[d], dinv[s] * dinv[d] * hw2[s]);
}

// ---------------- layer-2 epilogue: out += dinv^2*hw2 + b2 ----------------
__global__ void k_epi2(const float* __restrict__ hw2, const float* __restrict__ dinv,
                       const float* __restrict__ b2, float* __restrict__ out, int n) {
    int i = blockIdx.x * blockDim.x + threadIdx.x;
    if (i >= n) return;
    float di = dinv[i];
    out[i] += di * di * hw2[i] + b2[0];
}

extern "C" void kernel_launch(void* const* d_in, const int* in_sizes, int n_in,
                              void* d_out, int out_size, void* d_ws, size_t ws_size,
                              hipStream_t stream) {
    const float* X  = (const float*)d_in[0];
    const float* W1 = (const float*)d_in[1];
    const float* b1 = (const float*)d_in[2];
    const float* W2 = (const float*)d_in[3];
    const float* b2 = (const float*)d_in[4];
    const int*   ei = (const int*)d_in[5];

    const int n = in_sizes[0] / F;   // 100000
    const int e = in_sizes[5] / 2;   // 1.6M
    const int* src = ei;             // edge_index[0]
    const int* dst = ei + e;         // edge_index[1]

    // workspace layout
    char* ws = (char*)d_ws;
    size_t off = 0;
    auto alloc = [&](size_t bytes) -> void* {
        off = (off + 255) & ~(size_t)255;
        void* p = ws + off;
        off += bytes;
        return p;
    };
    unsigned* deg  = (unsigned*)alloc((size_t)n * 4);
    float*    dinv = (float*)   alloc((size_t)n * 4);
    float*    hw1  = (float*)   alloc((size_t)n * F * 4);
    float*    agg  = (float*)   alloc((size_t)n * F * 4);  // becomes h in-place
    float*    hw2  = (float*)   alloc((size_t)n * 4);
    float*    out  = (float*)d_out;

    const int B = 256;
    const long long nf = (long long)n * F;

    // degrees + normalization
    k_deg_init <<<(n + B - 1) / B, B, 0, stream>>>(deg, n);
    k_deg_count<<<(e + B - 1) / B, B, 0, stream>>>(dst, deg, e);
    k_dinv     <<<(n + B - 1) / B, B, 0, stream>>>(deg, dinv, n);

    // layer 1
    k_gemm1<<<n / 16, B, 0, stream>>>(X, W1, hw1);
    k_zero <<<(unsigned)((nf + B - 1) / B), B, 0, stream>>>(agg, nf);
    {
        long long tot = (long long)e * 32;
        k_scatter1<<<(unsigned)((tot + B - 1) / B), B, 0, stream>>>(hw1, dinv, src, dst, agg, e);
    }
    k_epi1<<<(unsigned)((nf + B - 1) / B), B, 0, stream>>>(hw1, dinv, b1, agg, nf);

    // layer 2
    k_matvec2 <<<(n + 7) / 8, B, 0, stream>>>(agg, W2, hw2, n);
    k_zero    <<<(n + B - 1) / B, B, 0, stream>>>(out, n);
    k_scatter2<<<(e + B - 1) / B, B, 0, stream>>>(hw2, dinv, src, dst, out, e);
    k_epi2    <<<(n + B - 1) / B, B, 0, stream>>>(hw2, dinv, b2, out, n);
}